// Attention_65876208386508
// MI455X (gfx1250) — compile-verified
//
#include <hip/hip_runtime.h>
#include <hip/hip_bf16.h>
#include <stdint.h>

// ---------------------------------------------------------------------------
// scores[b,s] = sum_d E[b,s,d] * h[b,d]    (b=64, s=2048, d=1024, fp32)
// Memory-bound: 512 MiB of E streamed once at 0.5 FLOP/byte -> ~23 us floor
// at 23.3 TB/s; needs only ~12 TFLOP/s fp32, so VALU FMAs suffice and WMMA
// (1 useful column of 16 on the slow f32 matrix path) would be pure waste.
// Strategy: saturate HBM with CDNA5 GLOBAL_LOAD_ASYNC_TO_LDS_B128 copies into
// wave-private, triple-buffered LDS tiles; two chunks of DMA kept in flight
// (s_wait_asynccnt <= 8) to cover HBM latency while FMAs consume chunk c.
// ---------------------------------------------------------------------------

#define AS3 __attribute__((address_space(3)))
#define AS1 __attribute__((address_space(1)))

// Match the builtin's parameter type exactly: GCC-vector int4.
typedef int v4i __attribute__((vector_size(16)));

#if __has_builtin(__builtin_amdgcn_global_load_async_to_lds_b128)
#define USE_ASYNC_LDS 1
#else
#define USE_ASYNC_LDS 0
#endif

// Integer round-trip casts: always-legal way to materialize addrspace pointers.
__device__ __forceinline__ AS3 v4i* lds_cast4(const void* p) {
    return (AS3 v4i*)(uint32_t)(uintptr_t)p;   // AS(3) ptrs are 32-bit LDS offsets
}
__device__ __forceinline__ AS1 v4i* glob_cast4(const void* p) {
    return (AS1 v4i*)(uintptr_t)p;
}

template <int N>
__device__ __forceinline__ void wait_async() {
    asm volatile("" ::: "memory");              // no LDS reads hoisted above the wait
#if __has_builtin(__builtin_amdgcn_s_wait_asynccnt)
    __builtin_amdgcn_s_wait_asynccnt(N);
#else
    asm volatile("s_wait_asynccnt %0" :: "n"(N));
#endif
    asm volatile("" ::: "memory");
}

// ---------------------------------------------------------------------------
// Async-DMA staged GEMV.
// Block = 256 threads (8 waves). Each wave owns 16 rows of one batch.
// K is processed in KC=32-float chunks, TRIPLE-buffered in LDS with the DMA
// running two chunks ahead of the FMAs (ASYNCcnt <= 8 keeps 2x4 copies live).
// Copy side:    4 x global_load_async_to_lds_b128 per chunk (coalesced rows).
// Compute side: lane L handles row (L&15); lanes L and L+16 split the chunk's
//               K-range, so the final reduction is a single shfl_xor(16).
// LDS rows padded by 16B (RSTRIDE=36 dwords) -> conflict-free b128 ds reads.
// Per-wave ASYNCcnt is the only sync needed: each wave stages only tiles it
// consumes itself, so there are no workgroup barriers in the main loop.
// ---------------------------------------------------------------------------
#if USE_ASYNC_LDS
__global__ __launch_bounds__(256) void gemv_async_kernel(
    const float* __restrict__ h, const float* __restrict__ E,
    float* __restrict__ out, int S) {
    constexpr int D       = 1024;
    constexpr int KC      = 32;           // floats per chunk per row
    constexpr int NCH     = D / KC;       // 32 chunks
    constexpr int RSTRIDE = KC + 4;       // dwords per LDS row (+16B pad)
    constexpr int WBUF    = 16 * RSTRIDE; // dwords per buffer per wave (=576)
    constexpr int NBUF    = 3;            // triple buffer, 2 chunks of lookahead

    __shared__ alignas(16) float smem[D + 8 * NBUF * WBUF];  // 4KB h + 54KB tiles

    const int b    = blockIdx.y;
    const int tid  = threadIdx.x;
    const int wave = tid >> 5;
    const int lane = tid & 31;

    // Stage h[b,:] into LDS once (256 threads x float4 = 4KB).
    ((float4*)smem)[tid] = ((const float4*)(h + (size_t)b * D))[tid];
    __syncthreads();

    const int    s0    = blockIdx.x * 128 + wave * 16;
    const size_t ebase = ((size_t)b * S + s0) * (size_t)D;    // in floats
    float*       wbuf  = smem + D + wave * (NBUF * WBUF);     // wave-private
    const float* hl    = smem;

    auto issue_chunk = [&](int c, int bi) {
        // 16 rows x 128B = 8 x 16B units/row; 128 units over 4 wave-issues.
#pragma unroll
        for (int i = 0; i < 4; ++i) {
            const int t = i * 32 + lane;
            const int row = t >> 3, unit = t & 7;
            const float* g = E + ebase + (size_t)row * D + c * KC + unit * 4;
            float*       l = wbuf + bi * WBUF + row * RSTRIDE + unit * 4;
            __builtin_amdgcn_global_load_async_to_lds_b128(
                glob_cast4(g), lds_cast4(l), /*offset=*/0, /*cpol=*/0);
        }
    };

    const int crow = lane & 15;
    const int half = lane >> 4;
    float acc = 0.0f;

    issue_chunk(0, 0);
    issue_chunk(1, 1);
    for (int c = 0; c < NCH; ++c) {
        if (c + 2 < NCH) {
            issue_chunk(c + 2, (c + 2) % NBUF);
            wait_async<8>();            // chunk c done; c+1, c+2 still in flight
        } else if (c + 1 < NCH) {
            wait_async<4>();            // chunk c done; c+1 still in flight
        } else {
            wait_async<0>();
        }
        const float* eb = wbuf + (c % NBUF) * WBUF + crow * RSTRIDE + half * 16;
        const float* hb = hl + c * KC + half * 16;
#pragma unroll
        for (int j = 0; j < 4; ++j) {
            float4 e  = *(const float4*)(eb + j * 4);
            float4 hh = *(const float4*)(hb + j * 4);
            acc += e.x * hh.x + e.y * hh.y + e.z * hh.z + e.w * hh.w;
        }
    }

    acc += __shfl_xor(acc, 16, 32);     // join the two K-halves of each row
    if (lane < 16) out[(size_t)b * S + s0 + crow] = acc;
}
#else  // !USE_ASYNC_LDS
// ---------------------------------------------------------------------------
// Fallback: direct-load streaming GEMV (bandwidth-optimal, no LDS round-trip).
// Wave per 8 rows; h held in 32 VGPRs/lane; wave32 shuffle-tree reduction.
// ---------------------------------------------------------------------------
__global__ __launch_bounds__(256) void gemv_base_kernel(
    const float* __restrict__ h, const float* __restrict__ E,
    float* __restrict__ out, int S) {
    constexpr int D = 1024;
    const int b    = blockIdx.y;
    const int wave = threadIdx.x >> 5;
    const int lane = threadIdx.x & 31;
    const int s0   = blockIdx.x * 64 + wave * 8;

    const float4* h4 = (const float4*)(h + (size_t)b * D);
    float4 hr[8];
#pragma unroll
    for (int i = 0; i < 8; ++i) hr[i] = h4[lane + i * 32];

    const float4* E4 = (const float4*)(E + ((size_t)b * S + s0) * (size_t)D);
#pragma unroll
    for (int r = 0; r < 8; ++r) {
        const float4* row = E4 + (size_t)r * (D / 4);
        float acc = 0.0f;
#pragma unroll
        for (int i = 0; i < 8; ++i) {
            float4 e = row[lane + i * 32];
            acc += e.x * hr[i].x + e.y * hr[i].y + e.z * hr[i].z + e.w * hr[i].w;
        }
#pragma unroll
        for (int off = 16; off > 0; off >>= 1) acc += __shfl_xor(acc, off, 32);
        if (lane == 0) out[(size_t)b * S + s0 + r] = acc;
    }
}
#endif  // USE_ASYNC_LDS

extern "C" void kernel_launch(void* const* d_in, const int* in_sizes, int n_in,
                              void* d_out, int out_size, void* d_ws, size_t ws_size,
                              hipStream_t stream) {
    const float* h   = (const float*)d_in[0];   // [B, D]
    const float* E   = (const float*)d_in[1];   // [B, S, D]
    float*       out = (float*)d_out;           // [B, 1, S] -> B*S floats

    const int D  = 1024;
    const int B  = in_sizes[0] / D;             // 64
    const int S  = in_sizes[1] / in_sizes[0];   // 2048

#if USE_ASYNC_LDS
    dim3 grid(S / 128, B);                      // 16 x 64 blocks
    gemv_async_kernel<<<grid, 256, 0, stream>>>(h, E, out, S);
#else
    dim3 grid(S / 64, B);                       // 32 x 64 blocks
    gemv_base_kernel<<<grid, 256, 0, stream>>>(h, E, out, S);
#endif
}